// Edgelayer_18365280157998
// MI455X (gfx1250) — compile-verified
//
#include <hip/hip_runtime.h>
#include <hip/hip_bf16.h>

typedef float v2f __attribute__((ext_vector_type(2)));
typedef float v8f __attribute__((ext_vector_type(8)));

#define NEG_SLOPE 0.2f
#define MAXSEG 512   // max supported edges per segment (expected max ~70 for random edges)

// LDS byte offset of a pointer into __shared__ memory (addrspacecast strips aperture)
__device__ __forceinline__ unsigned lds_offset_of(const void* p) {
    return (unsigned)(uintptr_t)(__attribute__((address_space(3))) const void*)p;
}

// ---------------------------------------------------------------------------
// Kernel 0: zero counts + cursor (must re-init every call; harness won't).
// ---------------------------------------------------------------------------
__global__ void zero_kernel(int* __restrict__ counts, int* __restrict__ cursor, int n) {
    int i = blockIdx.x * blockDim.x + threadIdx.x;
    int stride = gridDim.x * blockDim.x;
    for (; i < n; i += stride) { counts[i] = 0; cursor[i] = 0; }
}

// ---------------------------------------------------------------------------
// Kernel 1: s_src = x @ att[0:64], s_dst = x @ att[64:128] via
// V_WMMA_F32_16X16X4_F32. One wave32 per 16-row tile of x.
// A (16x4 f32): lanes 0-15 => M=0..15 with K={k0,k0+1}; lanes 16-31 => K={k0+2,k0+3}.
// B (4x16 f32): lane%16 = N column; only N=0 (att_src) and N=1 (att_dst) nonzero.
// D (16x16 f32): lane L, vgpr v => M = v + 8*(L>=16), N = L%16.
// ---------------------------------------------------------------------------
__global__ void gemv_wmma_kernel(const float* __restrict__ x,
                                 const float* __restrict__ att,
                                 float* __restrict__ s_src,
                                 float* __restrict__ s_dst,
                                 int ntiles) {
    int waveInBlock = threadIdx.x >> 5;
    int lane        = threadIdx.x & 31;
    int tile = blockIdx.x * (blockDim.x >> 5) + waveInBlock;
    if (tile >= ntiles) return;               // wave-uniform: EXEC stays all-1s

    int m     = lane & 15;
    int khalf = (lane >> 4) << 1;             // 0 or 2
    int n     = lane & 15;
    const float* xr = x + (tile * 16 + m) * 64;

    v8f c = {0.f, 0.f, 0.f, 0.f, 0.f, 0.f, 0.f, 0.f};
#pragma unroll
    for (int k0 = 0; k0 < 64; k0 += 4) {
        v2f a;
        a.x = xr[k0 + khalf];
        a.y = xr[k0 + khalf + 1];
        v2f b; b.x = 0.f; b.y = 0.f;
        if (n < 2) {                          // column 0 = att_src, column 1 = att_dst
            const float* ap = att + n * 64 + k0 + khalf;
            b.x = ap[0];
            b.y = ap[1];
        }
        c = __builtin_amdgcn_wmma_f32_16x16x4_f32(false, a, false, b,
                                                  (short)0, c, false, false);
    }
    int mbase = tile * 16 + ((lane >> 4) << 3);
    if (n == 0) {
#pragma unroll
        for (int v = 0; v < 8; ++v) s_src[mbase + v] = c[v];
    } else if (n == 1) {
#pragma unroll
        for (int v = 0; v < 8; ++v) s_dst[mbase + v] = c[v];
    }
}

// Scalar tail for rows not covered by 16-row tiles (none for N=50000).
__global__ void gemv_tail_kernel(const float* __restrict__ x,
                                 const float* __restrict__ att,
                                 float* __restrict__ s_src,
                                 float* __restrict__ s_dst,
                                 int startRow, int N) {
    int r = startRow + blockIdx.x * blockDim.x + threadIdx.x;
    if (r >= N) return;
    float a = 0.f, b = 0.f;
    for (int k = 0; k < 64; ++k) {
        float xv = x[r * 64 + k];
        a += xv * att[k];
        b += xv * att[64 + k];
    }
    s_src[r] = a;
    s_dst[r] = b;
}

// ---------------------------------------------------------------------------
// Kernel 2: histogram of edges per source node.
// ---------------------------------------------------------------------------
__global__ void count_kernel(const int* __restrict__ row, int* __restrict__ counts, int E) {
    int e = blockIdx.x * blockDim.x + threadIdx.x;
    if (e >= E) return;
    __builtin_prefetch(row + e + 4096, 0, 0);   // global_prefetch_b8
    atomicAdd(&counts[row[e]], 1);
}

// ---------------------------------------------------------------------------
// Kernel 3: single-workgroup exclusive scan of counts -> starts (N ~ 50000).
// wave32 shfl scan + 32-entry cross-wave scan: 4 barriers per 1024 chunk.
// ---------------------------------------------------------------------------
__global__ void scan_kernel(const int* __restrict__ counts, int* __restrict__ starts, int n) {
    __shared__ int wsum[32];
    __shared__ int running_s;
    int t    = threadIdx.x;
    int lane = t & 31;
    int wave = t >> 5;                 // 32 waves of 32 lanes
    if (t == 0) running_s = 0;
    __syncthreads();
    for (int base = 0; base < n; base += 1024) {
        int i = base + t;
        int v = (i < n) ? counts[i] : 0;
        int incl = v;                  // wave-level inclusive scan (5 shfl steps)
#pragma unroll
        for (int off = 1; off < 32; off <<= 1) {
            int y = __shfl_up(incl, off, 32);
            if (lane >= off) incl += y;
        }
        if (lane == 31) wsum[wave] = incl;
        __syncthreads();
        if (wave == 0) {               // scan the 32 wave sums
            int s = wsum[lane];
#pragma unroll
            for (int off = 1; off < 32; off <<= 1) {
                int y = __shfl_up(s, off, 32);
                if (lane >= off) s += y;
            }
            wsum[lane] = s;
        }
        __syncthreads();
        int waveOff = (wave > 0) ? wsum[wave - 1] : 0;
        int excl = incl - v + waveOff + running_s;
        if (i < n) starts[i] = excl;
        __syncthreads();
        if (t == 1023) running_s = excl + v;   // running + chunk total
        __syncthreads();
    }
}

// ---------------------------------------------------------------------------
// Kernel 4: compute edge weight, scatter into segment-contiguous gvals.
// ---------------------------------------------------------------------------
__global__ void scatter_kernel(const int* __restrict__ row, const int* __restrict__ col,
                               const float* __restrict__ s_src, const float* __restrict__ s_dst,
                               const int* __restrict__ starts, int* __restrict__ cursor,
                               float* __restrict__ gvals, int E) {
    int e = blockIdx.x * blockDim.x + threadIdx.x;
    if (e >= E) return;
    int r = row[e];
    float w = s_src[r] + s_dst[col[e]];
    w = (w > 0.f) ? w : NEG_SLOPE * w;
    int pos = starts[r] + atomicAdd(&cursor[r], 1);
    gvals[pos] = w;
}

// ---------------------------------------------------------------------------
// Kernel 5: per-segment sparsemax threshold tau. One wave32 per segment,
// 8 segments per 256-thread block. Segment values are brought into LDS with
// GLOBAL_LOAD_ASYNC_TO_LDS_B32 (ASYNCcnt path, no VGPR roundtrip), then a
// rank-sort in LDS (n^2/32 compares per lane), then lane 0 finds
// k = max{k : 1 + k*z_(k) > cumsum_k} and tau = (cs_k - 1)/k.
// ---------------------------------------------------------------------------
__global__ void seg_sparsemax_kernel(const float* __restrict__ gvals,
                                     const int* __restrict__ starts,
                                     const int* __restrict__ counts,
                                     float* __restrict__ tau, int nseg) {
    __shared__ float vals[8][MAXSEG];
    __shared__ float sorted[8][MAXSEG];
    int wave = threadIdx.x >> 5;
    int lane = threadIdx.x & 31;
    int seg  = blockIdx.x * 8 + wave;

    int len = 0, start = 0;
    if (seg < nseg) { len = counts[seg]; start = starts[seg]; }
    if (len > MAXSEG) len = MAXSEG;     // safety clamp; statistically unreachable here

    // async global -> LDS copy of this wave's segment (4B per lane per issue)
    for (int i = lane; i < len; i += 32) {
        unsigned ldsoff = lds_offset_of(&vals[wave][i]);
        unsigned long long ga = (unsigned long long)(uintptr_t)(gvals + start + i);
        asm volatile("global_load_async_to_lds_b32 %0, %1, off"
                     :: "v"(ldsoff), "v"(ga) : "memory");
    }
    asm volatile("s_wait_asynccnt 0x0" ::: "memory");  // this wave's copies done
    __syncthreads();

    for (int i = lane; i < len; i += 32) {
        float vi = vals[wave][i];
        int r = 0;
        for (int j = 0; j < len; ++j) {
            float vj = vals[wave][j];
            r += (vj > vi) || (vj == vi && j < i);   // stable descending rank
        }
        sorted[wave][r] = vi;
    }
    __syncthreads();

    if (lane == 0 && seg < nseg) {
        float cs = 0.f, cs_k = 0.f;
        int k = 0;
        for (int r2 = 0; r2 < len; ++r2) {
            float z = sorted[wave][r2];
            cs += z;
            if (1.0f + (float)(r2 + 1) * z > cs) { k = r2 + 1; cs_k = cs; }
        }
        tau[seg] = (k > 0) ? (cs_k - 1.0f) / (float)k : 0.0f;
    }
}

// ---------------------------------------------------------------------------
// Kernel 6: out[e] = max(leaky_relu(s_src[row]+s_dst[col]) - tau[row], 0)
// ---------------------------------------------------------------------------
__global__ void output_kernel(const int* __restrict__ row, const int* __restrict__ col,
                              const float* __restrict__ s_src, const float* __restrict__ s_dst,
                              const float* __restrict__ tau, float* __restrict__ out, int E) {
    int e = blockIdx.x * blockDim.x + threadIdx.x;
    if (e >= E) return;
    __builtin_prefetch(row + e + 4096, 0, 0);
    int r = row[e];
    float w = s_src[r] + s_dst[col[e]];
    w = (w > 0.f) ? w : NEG_SLOPE * w;
    out[e] = fmaxf(w - tau[r], 0.0f);
}

// ---------------------------------------------------------------------------
extern "C" void kernel_launch(void* const* d_in, const int* in_sizes, int n_in,
                              void* d_out, int out_size, void* d_ws, size_t ws_size,
                              hipStream_t stream) {
    const float* x          = (const float*)d_in[0];
    const int*   edge_index = (const int*)d_in[1];
    const float* att        = (const float*)d_in[2];

    const int D = 64;
    int N = in_sizes[0] / D;        // 50000
    int E = in_sizes[1] / 2;        // 1,600,000
    const int* row = edge_index;
    const int* col = edge_index + E;
    float* out = (float*)d_out;

    // Workspace layout (element counts rounded up to 256 for alignment)
    int Npad = (N + 255) & ~255;
    int Epad = (E + 255) & ~255;
    float* s_src  = (float*)d_ws;           // Npad
    float* s_dst  = s_src + Npad;           // Npad
    float* tau    = s_dst + Npad;           // Npad
    float* gvals  = tau + Npad;             // Epad
    int*   counts = (int*)(gvals + Epad);   // Npad
    int*   starts = counts + Npad;          // Npad
    int*   cursor = starts + Npad;          // Npad
    (void)ws_size; (void)n_in; (void)out_size;

    // 0) zero counts/cursor
    zero_kernel<<<64, 256, 0, stream>>>(counts, cursor, N);

    // 1) s_src/s_dst via WMMA (16 rows per wave, 8 waves per block)
    int ntiles = N / 16;
    if (ntiles > 0) {
        int blocks = (ntiles + 7) / 8;
        gemv_wmma_kernel<<<blocks, 256, 0, stream>>>(x, att, s_src, s_dst, ntiles);
    }
    int rem = N - ntiles * 16;
    if (rem > 0) {
        gemv_tail_kernel<<<1, 256, 0, stream>>>(x, att, s_src, s_dst, ntiles * 16, N);
    }

    // 2) histogram by source node
    count_kernel<<<(E + 255) / 256, 256, 0, stream>>>(row, counts, E);

    // 3) exclusive scan -> starts
    scan_kernel<<<1, 1024, 0, stream>>>(counts, starts, N);

    // 4) scatter weights into segment-contiguous buffer
    scatter_kernel<<<(E + 255) / 256, 256, 0, stream>>>(row, col, s_src, s_dst,
                                                        starts, cursor, gvals, E);

    // 5) per-segment sparsemax threshold (one wave32 per segment)
    seg_sparsemax_kernel<<<(N + 7) / 8, 256, 0, stream>>>(gvals, starts, counts, tau, N);

    // 6) final elementwise output
    output_kernel<<<(E + 255) / 256, 256, 0, stream>>>(row, col, s_src, s_dst, tau, out, E);
}